// PrototypeBank_82695300317837
// MI455X (gfx1250) — compile-verified
//
#include <hip/hip_runtime.h>
#include <math.h>
#include <stdint.h>

typedef __attribute__((ext_vector_type(2))) float v2f;
typedef __attribute__((ext_vector_type(8))) float v8f;

#define N_ROWS   131072
#define DDIM     512
#define KPROTO   64
#define WAVES    16
#define ROWS_PER_BLOCK (WAVES * 16)
#define PL_STRIDE 516   // padded f32 stride for prototype rows (bank-conflict free)
#define WL_STRIDE 68    // padded f32 stride for staged-w rows

#if defined(__AMDGCN__) && __has_builtin(__builtin_amdgcn_global_load_async_to_lds_b128)
#define HAS_ASYNC_LDS 1
typedef int v4i_b128 __attribute__((vector_size(16)));
__device__ __forceinline__ void async_copy_b128(const void* g, void* l) {
    // builtin expects (v4i AS(1)* gsrc, v4i AS(3)* lds, imm offset, imm cpol)
    __builtin_amdgcn_global_load_async_to_lds_b128(
        (__attribute__((address_space(1))) v4i_b128*)g,
        (__attribute__((address_space(3))) v4i_b128*)l,
        0, 0);
}
__device__ __forceinline__ void wait_async0() {
#if __has_builtin(__builtin_amdgcn_s_wait_asynccnt)
    __builtin_amdgcn_s_wait_asynccnt(0);
#else
    asm volatile("s_wait_asynccnt 0x0" ::: "memory");
#endif
}
#else
#define HAS_ASYNC_LDS 0
#endif

__global__ __launch_bounds__(512)
void proto_bank_kernel(const float* __restrict__ z,
                       const float* __restrict__ proto,
                       float* __restrict__ pattern,
                       float* __restrict__ wOut,
                       float* __restrict__ logitsOut)
{
    __shared__ float pl[KPROTO * PL_STRIDE];          // normalized prototypes
    __shared__ float wls[WAVES * 16 * WL_STRIDE];     // per-wave w' staging
    __shared__ float npArr[KPROTO];                   // max(||p||, eps)

    const int t = threadIdx.x;

    // ---- cooperative preload of raw prototypes into padded LDS ----
#if HAS_ASYNC_LDS
    {
        const float4* src = (const float4*)proto;
        #pragma unroll
        for (int i = 0; i < 16; ++i) {
            int g   = t + i * 512;        // float4 index (8192 total)
            int row = g >> 7;             // 128 float4 per proto row
            int c4  = g & 127;
            async_copy_b128(&src[g], &pl[row * PL_STRIDE + c4 * 4]);
        }
        wait_async0();
    }
#else
    {
        const float4* src = (const float4*)proto;
        #pragma unroll
        for (int i = 0; i < 16; ++i) {
            int g   = t + i * 512;
            int row = g >> 7;
            int c4  = g & 127;
            float4 v = src[g];
            *(float4*)&pl[row * PL_STRIDE + c4 * 4] = v;
        }
    }
#endif
    __syncthreads();

    // ---- normalize prototypes in LDS; record norms (threads 0..63) ----
    if (t < KPROTO) {
        float* row = &pl[t * PL_STRIDE];
        float ss = 0.f;
        for (int i = 0; i < DDIM; i += 4) {
            float4 v = *(float4*)&row[i];
            ss = fmaf(v.x, v.x, ss); ss = fmaf(v.y, v.y, ss);
            ss = fmaf(v.z, v.z, ss); ss = fmaf(v.w, v.w, ss);
        }
        float np = fmaxf(sqrtf(ss), 1e-12f);
        npArr[t] = np;
        float rn = 1.0f / np;
        for (int i = 0; i < DDIM; i += 4) {
            float4 v = *(float4*)&row[i];
            v.x *= rn; v.y *= rn; v.z *= rn; v.w *= rn;
            *(float4*)&row[i] = v;
        }
    }
    __syncthreads();

    const int lane = t & 31;
    const int wave = t >> 5;
    const int r    = lane & 15;
    const int hi   = lane >> 4;
    const int rowBase = blockIdx.x * ROWS_PER_BLOCK + wave * 16;

    // ---- matmul 1: u = z_tile(16xD) @ p_norm^T (Dx64), fused row-sumsq ----
    // K-permuted b128 scheme: per 8-wide K block, lane loads float4 at col k0+4*hi;
    // WMMA#1 contracts columns {k0,k0+1,k0+4,k0+5}, WMMA#2 the other four.
    v8f c0 = {}, c1 = {}, c2 = {}, c3 = {};
    float ss = 0.f;
    const float* zp = z + (size_t)(rowBase + r) * DDIM + 4 * hi;
    const float* b0 = &pl[(r     ) * PL_STRIDE + 4 * hi];
    const float* b1 = &pl[(r + 16) * PL_STRIDE + 4 * hi];
    const float* b2 = &pl[(r + 32) * PL_STRIDE + 4 * hi];
    const float* b3 = &pl[(r + 48) * PL_STRIDE + 4 * hi];

    #pragma unroll 2
    for (int k0 = 0; k0 < DDIM; k0 += 8) {
        float4 a4 = *(const float4*)(zp + k0);
        ss = fmaf(a4.x, a4.x, ss);
        ss = fmaf(a4.y, a4.y, ss);
        ss = fmaf(a4.z, a4.z, ss);
        ss = fmaf(a4.w, a4.w, ss);
        v2f aLo; aLo.x = a4.x; aLo.y = a4.y;
        v2f aHi; aHi.x = a4.z; aHi.y = a4.w;
        float4 q0 = *(const float4*)(b0 + k0);
        float4 q1 = *(const float4*)(b1 + k0);
        float4 q2 = *(const float4*)(b2 + k0);
        float4 q3 = *(const float4*)(b3 + k0);
        v2f bv;
        bv.x = q0.x; bv.y = q0.y;
        c0 = __builtin_amdgcn_wmma_f32_16x16x4_f32(false, aLo, false, bv, (short)0, c0, false, false);
        bv.x = q0.z; bv.y = q0.w;
        c0 = __builtin_amdgcn_wmma_f32_16x16x4_f32(false, aHi, false, bv, (short)0, c0, false, false);
        bv.x = q1.x; bv.y = q1.y;
        c1 = __builtin_amdgcn_wmma_f32_16x16x4_f32(false, aLo, false, bv, (short)0, c1, false, false);
        bv.x = q1.z; bv.y = q1.w;
        c1 = __builtin_amdgcn_wmma_f32_16x16x4_f32(false, aHi, false, bv, (short)0, c1, false, false);
        bv.x = q2.x; bv.y = q2.y;
        c2 = __builtin_amdgcn_wmma_f32_16x16x4_f32(false, aLo, false, bv, (short)0, c2, false, false);
        bv.x = q2.z; bv.y = q2.w;
        c2 = __builtin_amdgcn_wmma_f32_16x16x4_f32(false, aHi, false, bv, (short)0, c2, false, false);
        bv.x = q3.x; bv.y = q3.y;
        c3 = __builtin_amdgcn_wmma_f32_16x16x4_f32(false, aLo, false, bv, (short)0, c3, false, false);
        bv.x = q3.z; bv.y = q3.w;
        c3 = __builtin_amdgcn_wmma_f32_16x16x4_f32(false, aHi, false, bv, (short)0, c3, false, false);
    }

    // combine sumsq across the (l, l^16) pair -> full row sumsq at lane r
    ss += __shfl_xor(ss, 16, 32);
    // logits scale: 1 / (max(||z||, eps) * tau), tau = 0.2
    float sc = 5.0f / fmaxf(sqrtf(ss), 1e-12f);

    // broadcast per-row scales to the lanes holding that row in C-layout
    float rs[8];
    #pragma unroll
    for (int j = 0; j < 8; ++j) rs[j] = __shfl(sc, j + 8 * hi, 32);

    const float np0 = npArr[r];
    const float np1 = npArr[r + 16];
    const float np2 = npArr[r + 32];
    const float np3 = npArr[r + 48];

    // ---- scale -> softmax (half-wave shuffle reductions) -> outputs ----
    #pragma unroll
    for (int j = 0; j < 8; ++j) {
        float l0 = c0[j] * rs[j], l1 = c1[j] * rs[j];
        float l2 = c2[j] * rs[j], l3 = c3[j] * rs[j];
        float m = fmaxf(fmaxf(l0, l1), fmaxf(l2, l3));
        #pragma unroll
        for (int d = 1; d < 16; d <<= 1) m = fmaxf(m, __shfl_xor(m, d, 32));
        float e0 = __expf(l0 - m), e1 = __expf(l1 - m);
        float e2 = __expf(l2 - m), e3 = __expf(l3 - m);
        float s = e0 + e1 + e2 + e3;
        #pragma unroll
        for (int d = 1; d < 16; d <<= 1) s += __shfl_xor(s, d, 32);
        float inv = 1.0f / s;
        float w0 = e0 * inv, w1 = e1 * inv, w2 = e2 * inv, w3 = e3 * inv;

        int row = rowBase + j + 8 * hi;
        size_t ob = (size_t)row * KPROTO + r;
        logitsOut[ob] = l0; logitsOut[ob + 16] = l1;
        logitsOut[ob + 32] = l2; logitsOut[ob + 48] = l3;
        wOut[ob] = w0; wOut[ob + 16] = w1;
        wOut[ob + 32] = w2; wOut[ob + 48] = w3;

        // stage w' = w * ||p||  (reconstructs raw-prototype matmul)
        float* wr = &wls[(wave * 16 + j + 8 * hi) * WL_STRIDE + r];
        wr[0]  = w0 * np0; wr[16] = w1 * np1;
        wr[32] = w2 * np2; wr[48] = w3 * np3;
    }
    __syncthreads();   // cross-lane LDS transpose visibility

    // ---- matmul 2: pattern_tile = w'(16x64) @ p_norm(64x512) ----
    v2f aw[16];
    const float* wlr = &wls[(wave * 16 + r) * WL_STRIDE + 2 * hi];
    #pragma unroll
    for (int ks = 0; ks < 16; ++ks) aw[ks] = *(const v2f*)(wlr + 4 * ks);

    #pragma unroll 1
    for (int dt = 0; dt < 32; ++dt) {
        v8f acc = {};
        const float* bp = &pl[(2 * hi) * PL_STRIDE + 16 * dt + r];
        #pragma unroll
        for (int ks = 0; ks < 16; ++ks) {
            v2f b;
            b.x = bp[(4 * ks    ) * PL_STRIDE];
            b.y = bp[(4 * ks + 1) * PL_STRIDE];
            acc = __builtin_amdgcn_wmma_f32_16x16x4_f32(false, aw[ks], false, b, (short)0, acc, false, false);
        }
        #pragma unroll
        for (int j = 0; j < 8; ++j) {
            pattern[(size_t)(rowBase + j + 8 * hi) * DDIM + 16 * dt + r] = acc[j];
        }
    }
}

extern "C" void kernel_launch(void* const* d_in, const int* in_sizes, int n_in,
                              void* d_out, int out_size, void* d_ws, size_t ws_size,
                              hipStream_t stream) {
    (void)in_sizes; (void)n_in; (void)out_size; (void)d_ws; (void)ws_size;
    const float* z     = (const float*)d_in[0];   // [131072, 512]
    const float* proto = (const float*)d_in[1];   // [64, 512]
    float* out       = (float*)d_out;
    float* pattern   = out;                                   // [N, 512]
    float* wOut      = out + (size_t)N_ROWS * DDIM;           // [N, 64]
    float* logitsOut = wOut + (size_t)N_ROWS * KPROTO;        // [N, 64]

    dim3 grid(N_ROWS / ROWS_PER_BLOCK);   // 512 blocks
    dim3 block(512);                      // 16 waves of 32
    proto_bank_kernel<<<grid, block, 0, stream>>>(z, proto, pattern, wOut, logitsOut);
}